// RoIPool_9294309228861
// MI455X (gfx1250) — compile-verified
//
#include <hip/hip_runtime.h>
#include <hip/hip_bf16.h>
#include <math.h>

// Problem constants (from reference setup_inputs)
#define RB 4
#define RC 256
#define RH 50
#define RW 50
#define HWSZ (RH * RW)            // 2500
#define BSTRIDE (RC * HWSZ)       // 640000 floats per batch image
#define OUT_H 7
#define OUT_W 7
#define NBINS (OUT_H * OUT_W)     // 49
#define PER_ROI (RC * NBINS)      // 12544 floats per roi
#define SPATIAL_SCALE 0.0625f

#ifndef __has_builtin
#define __has_builtin(x) 0
#endif

#if __has_builtin(__builtin_amdgcn_global_store_async_from_lds_b128)
#define USE_ASYNC_STORE 1
typedef int v4i __attribute__((vector_size(16)));
typedef __attribute__((address_space(1))) v4i as1_v4i;   // global pointee
typedef __attribute__((address_space(3))) v4i as3_v4i;   // LDS pointee
#else
#define USE_ASYNC_STORE 0
#endif

// ---------------------------------------------------------------------------
// Transpose NCHW -> NHWC so channel gathers become coalesced 1KB row reads.
// Block (32,8): 32x32 tile per block, 4-row loop, padded LDS tile.
// ---------------------------------------------------------------------------
__global__ void nchw_to_nhwc(const float* __restrict__ in, float* __restrict__ out) {
    __shared__ float tile[32][33];
    const int tx = threadIdx.x;           // 0..31
    const int ty = threadIdx.y;           // 0..7
    const int hw0 = blockIdx.x * 32;      // spatial tile origin (flat h*W+w)
    const int c0  = blockIdx.y * 32;      // channel tile origin
    const int b   = blockIdx.z;

    const float* inb  = in  + (size_t)b * BSTRIDE;
    float*       outb = out + (size_t)b * BSTRIDE;

#pragma unroll
    for (int j = 0; j < 4; ++j) {
        const int c  = c0 + ty + j * 8;
        const int hw = hw0 + tx;
        if (hw < HWSZ)
            tile[ty + j * 8][tx] = inb[(size_t)c * HWSZ + hw];
    }
    __syncthreads();
#pragma unroll
    for (int j = 0; j < 4; ++j) {
        const int hw = hw0 + ty + j * 8;
        const int c  = c0 + tx;
        if (hw < HWSZ)
            outb[(size_t)hw * RC + c] = tile[tx][ty + j * 8];
    }
}

// ---------------------------------------------------------------------------
// RoIPool: one 256-thread block per roi; thread == channel.
// TRANSPOSED=true reads NHWC scratch (coalesced); false reads NCHW directly.
// Results staged in LDS in exact output order, drained with async b128 stores.
// ---------------------------------------------------------------------------
template <bool TRANSPOSED>
__global__ void roipool_kernel(const float* __restrict__ feat,
                               const float* __restrict__ rois,
                               float* __restrict__ out) {
    __shared__ int4 sb[NBINS];
    __shared__ __align__(16) float lres[PER_ROI];   // [c][bin] == output order

    const int k   = blockIdx.x;
    const int tid = threadIdx.x;                    // channel
    const float* roi = rois + (size_t)k * 5;

    const int b  = (int)roi[0];
    const int x1 = (int)rintf(roi[1] * SPATIAL_SCALE);   // round-half-even
    const int y1 = (int)rintf(roi[2] * SPATIAL_SCALE);
    const int x2 = (int)rintf(roi[3] * SPATIAL_SCALE);
    const int y2 = (int)rintf(roi[4] * SPATIAL_SCALE);
    const float roi_w = (float)max(x2 - x1 + 1, 1);
    const float roi_h = (float)max(y2 - y1 + 1, 1);
    const float bin_w = roi_w / (float)OUT_W;
    const float bin_h = roi_h / (float)OUT_H;

    if (tid < NBINS) {
        const int ph = tid / OUT_W;
        const int pw = tid % OUT_W;
        int hs = (int)floorf((float)ph * bin_h) + y1;
        int he = (int)ceilf(((float)ph + 1.0f) * bin_h) + y1;
        int ws = (int)floorf((float)pw * bin_w) + x1;
        int we = (int)ceilf(((float)pw + 1.0f) * bin_w) + x1;
        hs = min(max(hs, 0), RH);
        he = min(max(he, 0), RH);
        ws = min(max(ws, 0), RW);
        we = min(max(we, 0), RW);
        sb[tid] = make_int4(hs, he, ws, we);
    }
    __syncthreads();

    const int c = tid;
    const float* fb;
    if (TRANSPOSED) fb = feat + (size_t)b * BSTRIDE;                   // [hw][c]
    else            fb = feat + ((size_t)b * RC + c) * HWSZ;           // [c][hw]

    for (int bin = 0; bin < NBINS; ++bin) {
        const int4 bb = sb[bin];   // x=hs, y=he, z=ws, w=we (uniform broadcast)
        float m = 0.0f;
        if (bb.x < bb.y && bb.z < bb.w) {
            m = -INFINITY;
            for (int h = bb.x; h < bb.y; ++h) {
                if (TRANSPOSED) {
                    const float* row = fb + ((size_t)(h * RW + bb.z)) * RC + c;
                    if (h + 1 < bb.y)   // pull next cell row toward L0/L2
                        __builtin_prefetch(fb + ((size_t)((h + 1) * RW + bb.z)) * RC + c, 0, 1);
                    for (int w = bb.z; w < bb.w; ++w) {
                        m = fmaxf(m, *row);
                        row += RC;
                    }
                } else {
                    const float* row = fb + h * RW;
                    for (int w = bb.z; w < bb.w; ++w)
                        m = fmaxf(m, row[w]);
                }
            }
        }
        lres[c * NBINS + bin] = m;   // stride 49: conflict-free (gcd(49,64)=1)
    }
    __syncthreads();

    // Drain LDS -> global. LDS image is byte-identical to out[k,:,:,:].
    float* outk = out + (size_t)k * PER_ROI;
#if USE_ASYNC_STORE
    // 3136 float4 chunks; 16B-aligned on both sides (PER_ROI*4 % 16 == 0).
    for (int i = tid; i < PER_ROI / 4; i += 256) {
        __builtin_amdgcn_global_store_async_from_lds_b128(
            (as1_v4i*)(outk + (size_t)i * 4),
            (as3_v4i*)(lres + (size_t)i * 4),
            0, 0);
    }
#if __has_builtin(__builtin_amdgcn_s_wait_asynccnt)
    __builtin_amdgcn_s_wait_asynccnt(0);
#else
    asm volatile("s_wait_asynccnt 0" ::: "memory");
#endif
#else
    for (int i = tid; i < PER_ROI; i += 256)
        outk[i] = lres[i];
#endif
}

extern "C" void kernel_launch(void* const* d_in, const int* in_sizes, int n_in,
                              void* d_out, int out_size, void* d_ws, size_t ws_size,
                              hipStream_t stream) {
    const float* features = (const float*)d_in[0];
    const float* rois     = (const float*)d_in[1];
    float* out            = (float*)d_out;
    const int K = in_sizes[1] / 5;   // 2000

    const size_t need = (size_t)RB * BSTRIDE * sizeof(float);  // 10.24 MB NHWC scratch
    if (ws_size >= need) {
        float* featT = (float*)d_ws;
        dim3 tgrid((HWSZ + 31) / 32, RC / 32, RB);
        nchw_to_nhwc<<<tgrid, dim3(32, 8), 0, stream>>>(features, featT);
        roipool_kernel<true><<<dim3(K), dim3(256), 0, stream>>>(featT, rois, out);
    } else {
        roipool_kernel<false><<<dim3(K), dim3(256), 0, stream>>>(features, rois, out);
    }
}